// NeuralODE_64733747085918
// MI455X (gfx1250) — compile-verified
//
#include <hip/hip_runtime.h>
#include <stdint.h>

// ---------------- CDNA5 async LDS->global store helpers ----------------
#define ASL __attribute__((address_space(3)))

typedef float v2f __attribute__((ext_vector_type(2)));
typedef float v8f __attribute__((ext_vector_type(8)));

__device__ __forceinline__ void async_store_b64(unsigned long long gaddr, unsigned ldsoff) {
    // VGLOBAL: global_store_async_from_lds_b64 vaddr(64b pair), vsrc(LDS byte addr), off
    // Inline asm bypasses the clang builtin prototype (pointer typing differs across
    // toolchains); tracked by ASYNCcnt.
    asm volatile("global_store_async_from_lds_b64 %0, %1, off"
                 :: "v"(gaddr), "v"(ldsoff) : "memory");
}

__device__ __forceinline__ void wait_async0() {
#if __has_builtin(__builtin_amdgcn_s_wait_asynccnt)
    __builtin_amdgcn_s_wait_asynccnt(0);
#else
    asm volatile("s_wait_asynccnt 0x0" ::: "memory");
#endif
}

__device__ __forceinline__ void wait_ds0() {
    asm volatile("s_wait_dscnt 0x0" ::: "memory");
}

// ---------------- configuration ----------------
#define NT    256          // threads per block (8 wave32)
#define NW    (NT / 32)
#define CH    26           // time-slots staged in LDS per flush chunk
#define CHLAST 23          // final partial chunk: 101 = 3*26 + 23
#define PITCH 54           // dwords per trajectory row in LDS (CH*2 + 2 pad, reduces bank conflict)

// Drain one wave's 32 trajectories x COUNT slots from LDS to global via async DMA.
// Lane-remapped so consecutive lanes write consecutive 8B addresses (runs of COUNT
// float2 per trajectory -> ~200B contiguous bursts instead of scattered 8B stores).
// COUNT is a template constant so i/COUNT compiles to a magic-multiply.
template <int COUNT>
__device__ __forceinline__ void wave_flush(unsigned long long outBase, int waveB,
                                           int chunkBase, unsigned ldsWave, int lane) {
    wait_ds0();   // make staged ds_store data visible to the async engine
    const int E = 32 * COUNT;            // multiple of 32 -> full EXEC every iteration
    for (int i = lane; i < E; i += 32) {
        unsigned bl = (unsigned)i / (unsigned)COUNT;
        unsigned j  = (unsigned)i - bl * (unsigned)COUNT;
        unsigned long long g = outBase +
            ((unsigned long long)(((unsigned)waveB + bl) * 101u + (unsigned)chunkBase + j)) * 8ull;
        unsigned l = ldsWave + (bl * PITCH + j * 2u) * 4u;
        async_store_b64(g, l);
    }
}

// ---------------- main Tsit5 kernel ----------------
__global__ __launch_bounds__(NT) void NeuralODE_tsit5_kernel(
        const float* __restrict__ x0, const float* __restrict__ te,
        const float* __restrict__ Mp, float* __restrict__ out, int B) {
    __shared__ float buf[NW][32][PITCH];   // 55296 B

    const int tid  = threadIdx.x;
    const int lane = tid & 31;
    const int wave = tid >> 5;
    const int b    = blockIdx.x * NT + tid;
    if (b >= B) return;

    const float m00 = Mp[0], m01 = Mp[1], m10 = Mp[2], m11 = Mp[3];

    float y0 = x0[2 * b], y1 = x0[2 * b + 1];
    float dt = 1e-3f;
    int   ns = 0;

    const unsigned ldsWave =
        (unsigned)(unsigned long long)(ASL char*)&buf[wave][0][0];
    const int waveB = blockIdx.x * NT + wave * 32;
    const unsigned long long outBase = (unsigned long long)(uintptr_t)out;

    // time index 0: the initial condition
    *(float2*)&buf[wave][lane][0] = make_float2(y0, y1);
    int slot = 1, chunkBase = 0;

    for (int s = 0; s < 100; ++s) {
        if (slot == CH) {   // block-uniform: flush filled chunk, DMA overlaps next interval
            wave_flush<CH>(outBase, waveB, chunkBase, ldsWave, lane);
            chunkBase += CH;
            slot = 0;
        }

        const float t0v = te[s], t1v = te[s + 1];
        float t = t0v;
        const float t1m = t1v - 1e-8f;

#pragma unroll 1
        for (int it = 0; it < 16; ++it) {
            const bool notdone = t < t1m;
            if (__ballot((int)notdone) == 0ull) break;   // whole wave finished interval

            float h = fminf(dt, t1v - t);
            h = fmaxf(h, 1e-10f);

            const float k1x = m00 * y0 + m01 * y1;
            const float k1y = m10 * y0 + m11 * y1;

            float u0 = y0 + h * (0.161f * k1x);
            float u1 = y1 + h * (0.161f * k1y);
            const float k2x = m00 * u0 + m01 * u1;
            const float k2y = m10 * u0 + m11 * u1;

            u0 = y0 + h * (-0.008480655492356989f * k1x + 0.335480655492357f * k2x);
            u1 = y1 + h * (-0.008480655492356989f * k1y + 0.335480655492357f * k2y);
            const float k3x = m00 * u0 + m01 * u1;
            const float k3y = m10 * u0 + m11 * u1;

            u0 = y0 + h * (2.8971530571054935f * k1x - 6.359448489975075f * k2x + 4.3622954328695815f * k3x);
            u1 = y1 + h * (2.8971530571054935f * k1y - 6.359448489975075f * k2y + 4.3622954328695815f * k3y);
            const float k4x = m00 * u0 + m01 * u1;
            const float k4y = m10 * u0 + m11 * u1;

            u0 = y0 + h * (5.325864828439257f * k1x - 11.748883564062828f * k2x + 7.4955393428898365f * k3x - 0.09249506636175525f * k4x);
            u1 = y1 + h * (5.325864828439257f * k1y - 11.748883564062828f * k2y + 7.4955393428898365f * k3y - 0.09249506636175525f * k4y);
            const float k5x = m00 * u0 + m01 * u1;
            const float k5y = m10 * u0 + m11 * u1;

            u0 = y0 + h * (5.86145544294642f * k1x - 12.92096931784711f * k2x + 8.159367898576159f * k3x - 0.071584973281401f * k4x - 0.028269050394068383f * k5x);
            u1 = y1 + h * (5.86145544294642f * k1y - 12.92096931784711f * k2y + 8.159367898576159f * k3y - 0.071584973281401f * k4y - 0.028269050394068383f * k5y);
            const float k6x = m00 * u0 + m01 * u1;
            const float k6y = m10 * u0 + m11 * u1;

            const float yn0 = y0 + h * (0.09646076681806523f * k1x + 0.01f * k2x + 0.4798896504144996f * k3x + 1.379008574103742f * k4x - 3.290069515436081f * k5x + 2.324710524099774f * k6x);
            const float yn1 = y1 + h * (0.09646076681806523f * k1y + 0.01f * k2y + 0.4798896504144996f * k3y + 1.379008574103742f * k4y - 3.290069515436081f * k5y + 2.324710524099774f * k6y);

            const float k7x = m00 * yn0 + m01 * yn1;
            const float k7y = m10 * yn0 + m11 * yn1;

            const float e0 = h * (-0.001780011052225777f * k1x - 0.0008164344596567469f * k2x + 0.007880878010261995f * k3x - 0.1447110071732629f * k4x + 0.5823571654525552f * k5x - 0.45808210592918697f * k6x + 0.015151515151515152f * k7x);
            const float e1 = h * (-0.001780011052225777f * k1y - 0.0008164344596567469f * k2y + 0.007880878010261995f * k3y - 0.1447110071732629f * k4y + 0.5823571654525552f * k5y - 0.45808210592918697f * k6y + 0.015151515151515152f * k7y);

            const float sc0 = 1e-6f + 1e-3f * fmaxf(fabsf(y0), fabsf(yn0));
            const float sc1 = 1e-6f + 1e-3f * fmaxf(fabsf(y1), fabsf(yn1));
            const float r0 = e0 / sc0, r1 = e1 / sc1;
            const float enorm = sqrtf(0.5f * (r0 * r0 + r1 * r1));

            const bool accept = (enorm <= 1.0f) && notdone;
            // (enorm+1e-12)^-0.2 with strictly positive base: log2/exp2 hit the
            // TRANS pipe (v_log_f32/v_exp_f32) and co-execute with VALU.
            const float factor =
                fminf(fmaxf(0.9f * exp2f(-0.2f * log2f(enorm + 1e-12f)), 0.2f), 10.0f);

            t  = accept ? (t + h) : t;
            y0 = accept ? yn0 : y0;
            y1 = accept ? yn1 : y1;
            dt = notdone ? (h * factor) : dt;
            ns += notdone ? 1 : 0;
        }

        if (slot == 0 && chunkBase != 0) wait_async0();   // DMA of prev chunk done before reuse
        *(float2*)&buf[wave][lane][2 * slot] = make_float2(y0, y1);
        ++slot;
    }

    // final partial chunk (time indices 78..100 -> 23 slots, compile-time constant)
    wave_flush<CHLAST>(outBase, waveB, chunkBase, ldsWave, lane);

    // ---- ns wave-reduction on the matrix pipe (once per wave, EXEC all-ones here) ----
    // A-VGPR0 = per-lane ns (K0 slot for lanes 0-15, K2 slot for lanes 16-31), A-VGPR1 = 0,
    // B = all-ones  =>  D[m][n] = ns[m] + ns[m+16].  Per-lane sum of the 8 D VGPRs gives
    // lane0 = sum(rows 0..7), lane16 = sum(rows 8..15); one shfl finishes the wave sum.
    // ns <= 1600 per lane, wave total <= 51200 < 2^24: exact in f32.
    {
        v2f a;  a[0] = (float)ns; a[1] = 0.0f;
        v2f o;  o[0] = 1.0f;      o[1] = 1.0f;
        v8f c = {};
        v8f d = __builtin_amdgcn_wmma_f32_16x16x4_f32(false, a, false, o,
                                                      (short)0, c, false, false);
        float v = ((d[0] + d[1]) + (d[2] + d[3])) + ((d[4] + d[5]) + (d[6] + d[7]));
        v += __shfl_down(v, 16, 32);
        if (lane == 0) atomicAdd((int*)(out + (size_t)B * 202), (int)v);
    }
}

__global__ void NeuralODE_init_cnt(float* out, unsigned long long idx) {
    if (threadIdx.x == 0 && blockIdx.x == 0) *((int*)(out + idx)) = 0;
}

extern "C" void kernel_launch(void* const* d_in, const int* in_sizes, int n_in,
                              void* d_out, int out_size, void* d_ws, size_t ws_size,
                              hipStream_t stream) {
    const float* x0 = (const float*)d_in[0];   // (B,2)
    const float* te = (const float*)d_in[1];   // (101,)
    const float* M  = (const float*)d_in[2];   // (2,2)
    float* out = (float*)d_out;                // trajs (B,101,2) then int32 step-sum
    const int B = in_sizes[0] / 2;

    NeuralODE_init_cnt<<<1, 1, 0, stream>>>(out, (unsigned long long)B * 202ull);
    NeuralODE_tsit5_kernel<<<(B + NT - 1) / NT, NT, 0, stream>>>(x0, te, M, out, B);
}